// OpMessagePassing_42666205119406
// MI455X (gfx1250) — compile-verified
//
#include <hip/hip_runtime.h>

typedef __attribute__((ext_vector_type(2))) float v2f;
typedef __attribute__((ext_vector_type(8))) float v8f;

#define BATCH 16
#define NN    256
#define DD    64

// One wave computes a 16(i) x 64(d) block of out[b, :, k, :] for a fixed k,
// accumulating over j = 0..255 with V_WMMA_F32_16X16X4_F32 (4 tiles of 16 cols).
__global__ __launch_bounds__(256)
void mamamm_wmma_kernel(const float* __restrict__ A_data,
                        const int*   __restrict__ A_mask,
                        const float* __restrict__ B_data,
                        const int*   __restrict__ B_mask,
                        const int*   __restrict__ tar_mask,
                        float*       __restrict__ out)
{
    const int lane = threadIdx.x & 31;
    const int wave = threadIdx.x >> 5;
    const int m    = lane & 15;   // column within 16-wide tile / row within i-tile
    const int hi   = lane >> 4;   // lane half selects K pair (A/B) and M half (C/D)

    // unit id: i-tile fastest so waves sharing B[b,:,k,:] are adjacent (L2 reuse)
    const unsigned unit = blockIdx.x * 8u + wave;   // [0, 65536)
    const int it = unit & 15;                       // i tile (16 rows)
    const int k  = (unit >> 4) & 255;               // target node column
    const int b  = unit >> 12;                      // batch

    const int i0 = it * 16;

    // This lane's A row (row m of the i-tile)
    const float* Arow  = A_data + (size_t)(b * NN + i0 + m) * NN;
    const int*   AMrow = A_mask + (size_t)(b * NN + i0 + m) * NN;

    const size_t bBase = (size_t)b * NN * NN;       // base of (b, j, k) plane

    v8f c0 = {}, c1 = {}, c2 = {}, c3 = {};

    #pragma unroll 2
    for (int j = 0; j < NN; j += 4) {
        // K rows handled by this lane: r0 = j + 2*hi, r1 = r0 + 1
        const int r0 = j + 2 * hi;
        const int r1 = r0 + 1;

        // ---- A fragment (16x4 f32): VGPR0 = A[m][r0], VGPR1 = A[m][r1] ----
        const float av0 = Arow[r0];
        const float av1 = Arow[r1];
        const int   am0 = AMrow[r0];
        const int   am1 = AMrow[r1];
        v2f a;
        a.x = am0 ? av0 : 0.0f;
        a.y = am1 ? av1 : 0.0f;

        // ---- B rows: B[b, r, k, 0..63]; mask B_mask[b, r, k] broadcast over d ----
        const float* Brow0 = B_data + (bBase + (size_t)r0 * NN + k) * DD;
        const float* Brow1 = B_data + (bBase + (size_t)r1 * NN + k) * DD;

        // Prefetch next j-step's B rows (speculative; safe past end)
        __builtin_prefetch(Brow0 + (size_t)4 * NN * DD, 0, 1);
        __builtin_prefetch(Brow1 + (size_t)4 * NN * DD, 0, 1);

        // Unconditional loads; mask applied as a multiplier so the compiler
        // cannot sink the loads into predicated EXEC regions.
        const float u0 = Brow0[m],      w0 = Brow1[m];
        const float u1 = Brow0[m + 16], w1 = Brow1[m + 16];
        const float u2 = Brow0[m + 32], w2 = Brow1[m + 32];
        const float u3 = Brow0[m + 48], w3 = Brow1[m + 48];

        const float f0 = B_mask[bBase + (size_t)r0 * NN + k] ? 1.0f : 0.0f;
        const float f1 = B_mask[bBase + (size_t)r1 * NN + k] ? 1.0f : 0.0f;

        v2f b0, b1, b2, b3;
        b0.x = u0 * f0;  b0.y = w0 * f1;
        b1.x = u1 * f0;  b1.y = w1 * f1;
        b2.x = u2 * f0;  b2.y = w2 * f1;
        b3.x = u3 * f0;  b3.y = w3 * f1;

        // D = A x B + C  (fp32, 16x16x4)
        c0 = __builtin_amdgcn_wmma_f32_16x16x4_f32(false, a, false, b0, (short)0, c0, false, false);
        c1 = __builtin_amdgcn_wmma_f32_16x16x4_f32(false, a, false, b1, (short)0, c1, false, false);
        c2 = __builtin_amdgcn_wmma_f32_16x16x4_f32(false, a, false, b2, (short)0, c2, false, false);
        c3 = __builtin_amdgcn_wmma_f32_16x16x4_f32(false, a, false, b3, (short)0, c3, false, false);
    }

    // ---- store, applying tar_mask[b, i, k] (scalar per output row) ----
    #pragma unroll
    for (int r = 0; r < 8; ++r) {
        const int i = i0 + r + 8 * hi;                      // C/D layout: M = r + 8*hi
        const size_t rowIdx = (size_t)(b * NN + i) * NN + k;
        const float tm = tar_mask[rowIdx] ? 1.0f : 0.0f;
        float* orow = out + rowIdx * DD;
        orow[m]      = c0[r] * tm;
        orow[m + 16] = c1[r] * tm;
        orow[m + 32] = c2[r] * tm;
        orow[m + 48] = c3[r] * tm;
    }
}

extern "C" void kernel_launch(void* const* d_in, const int* in_sizes, int n_in,
                              void* d_out, int out_size, void* d_ws, size_t ws_size,
                              hipStream_t stream) {
    const float* A_data   = (const float*)d_in[0];
    const int*   A_mask   = (const int*)  d_in[1];
    const float* B_data   = (const float*)d_in[2];
    const int*   B_mask   = (const int*)  d_in[3];
    const int*   tar_mask = (const int*)  d_in[4];
    float* out = (float*)d_out;

    // 16 batches * 16 i-tiles * 256 k = 65536 wave-units, 8 waves per block
    dim3 grid(BATCH * (NN / 16) * NN / 8);   // 8192
    dim3 block(256);
    hipLaunchKernelGGL(mamamm_wmma_kernel, grid, block, 0, stream,
                       A_data, A_mask, B_data, B_mask, tar_mask, out);
}